// MTopDivYXLoss_26482768347684
// MI455X (gfx1250) — compile-verified
//
#include <hip/hip_runtime.h>
#include <hip/hip_bf16.h>

typedef float v2f __attribute__((ext_vector_type(2)));
typedef float v8f __attribute__((ext_vector_type(8)));

#define NA 4096
#define NB 4096
#define DIMK 128

// ---------------------------------------------------------------------------
// Kernel 1: squared norms, one wave per point (coalesced 512B per wave read)
// ---------------------------------------------------------------------------
__global__ __launch_bounds__(256) void sqnorm_kernel(const float* __restrict__ src,
                                                     float* __restrict__ sq, int npts) {
    int wid  = blockIdx.x * (blockDim.x >> 5) + (threadIdx.x >> 5);
    int lane = threadIdx.x & 31;
    if (wid >= npts) return;
    const float4* row = (const float4*)(src + (size_t)wid * DIMK);
    float4 v = row[lane];                       // 32 lanes x 4 floats = 128
    float  s = v.x * v.x + v.y * v.y + v.z * v.z + v.w * v.w;
    for (int off = 16; off > 0; off >>= 1) s += __shfl_down(s, off, 32);
    if (lane == 0) sq[wid] = s;
}

// ---------------------------------------------------------------------------
// Kernel 2: distance tile GEMM via V_WMMA_F32_16X16X4_F32.
// Block = 256 threads (8 waves), macro-tile 128x128, Kc = 32 LDS staging.
// Wave w owns subtile-row w (16 M-rows) x 8 N-subtiles -> 8 x v8f accumulators.
// Epilogue: D = sqrt(max(sqA[i] + sqB[j] - 2*dot, 0)).
// ---------------------------------------------------------------------------
#define TM 128
#define TN 128
#define KC 32
#define LDSPAD 36   // 32 + 4 floats padding: lanes stride 4 banks -> conflict-free b64 reads

__global__ __launch_bounds__(256) void dist_tile_kernel(const float* __restrict__ A,
                                                        const float* __restrict__ B,
                                                        const float* __restrict__ sqA,
                                                        const float* __restrict__ sqB,
                                                        float* __restrict__ Dout) {
    __shared__ float As[TM][LDSPAD];
    __shared__ float Bs[TN][LDSPAD];

    const int tid  = threadIdx.x;
    const int lane = tid & 31;
    const int wave = tid >> 5;
    const int bm   = blockIdx.y;
    const int bn   = blockIdx.x;

    v8f acc[8];
#pragma unroll
    for (int n = 0; n < 8; n++)
#pragma unroll
        for (int r = 0; r < 8; r++) acc[n][r] = 0.0f;

    const int half = lane >> 4;        // 0: K pair {0,1}, 1: K pair {2,3}
    const int ln16 = lane & 15;

    for (int kc = 0; kc < DIMK; kc += KC) {
        // Stage A and B tiles: 128 rows x 32 floats each = 1024 float4s, 4 per thread.
#pragma unroll
        for (int v = 0; v < 4; v++) {
            int f4  = tid + v * 256;       // 0..1023
            int row = f4 >> 3;             // /8 float4s per row
            int c4  = (f4 & 7) << 2;       // float offset within row
            float4 da = *(const float4*)(A + (size_t)(bm * TM + row) * DIMK + kc + c4);
            float4 db = *(const float4*)(B + (size_t)(bn * TN + row) * DIMK + kc + c4);
            As[row][c4 + 0] = da.x; As[row][c4 + 1] = da.y;
            As[row][c4 + 2] = da.z; As[row][c4 + 3] = da.w;
            Bs[row][c4 + 0] = db.x; Bs[row][c4 + 1] = db.y;
            Bs[row][c4 + 2] = db.z; Bs[row][c4 + 3] = db.w;
        }
        __syncthreads();

#pragma unroll
        for (int k0 = 0; k0 < KC; k0 += 4) {
            const int kk = k0 + (half << 1);
            v2f a;
            a.x = As[wave * 16 + ln16][kk];
            a.y = As[wave * 16 + ln16][kk + 1];
#pragma unroll
            for (int n = 0; n < 8; n++) {
                v2f b;
                b.x = Bs[n * 16 + ln16][kk];
                b.y = Bs[n * 16 + ln16][kk + 1];
                acc[n] = __builtin_amdgcn_wmma_f32_16x16x4_f32(
                    false, a, false, b, (short)0, acc[n], false, false);
            }
        }
        __syncthreads();
    }

    // Epilogue: C layout — VGPR r, lanes 0-15: M=r, N=lane; lanes 16-31: M=r+8, N=lane-16
    const int gm_base = bm * TM + wave * 16 + half * 8;
    float sa[8];
#pragma unroll
    for (int r = 0; r < 8; r++) sa[r] = sqA[gm_base + r];

#pragma unroll
    for (int n = 0; n < 8; n++) {
        const int gcol = bn * TN + n * 16 + ln16;
        const float sb = sqB[gcol];
#pragma unroll
        for (int r = 0; r < 8; r++) {
            const int grow = gm_base + r;
            float d2 = sa[r] + sb - 2.0f * acc[n][r];
            Dout[(size_t)grow * NB + gcol] = sqrtf(fmaxf(d2, 0.0f));
        }
    }
}

// ---------------------------------------------------------------------------
// Kernel 3a/3b: key[j] = min_i AB[i][j]  (contract A-cluster to supernode A*)
// Distances >= 0, so IEEE float bits compare monotonically as u32.
// ---------------------------------------------------------------------------
__global__ void keyinit_kernel(unsigned int* __restrict__ keybits) {
    keybits[blockIdx.x * 256 + threadIdx.x] = 0x7F800000u;   // +inf
}

__global__ __launch_bounds__(256) void colmin_kernel(const float* __restrict__ AB,
                                                     unsigned int* __restrict__ keybits) {
    const int col = blockIdx.x * 256 + threadIdx.x;
    const int r0  = blockIdx.y * 256;
    float m = __builtin_inff();
    for (int r = 0; r < 256; r++)
        m = fminf(m, AB[(size_t)(r0 + r) * NB + col]);
    atomicMin(&keybits[col], __float_as_uint(m));
}

// ---------------------------------------------------------------------------
// Kernel 4: Prim over the contracted graph {A*, B0..B4095}: 4096 serial picks.
// Single 1024-thread workgroup; key/visited LDS-resident; first-index tie-break
// to match jnp.argmin semantics.
// ---------------------------------------------------------------------------
__global__ __launch_bounds__(1024) void prim_kernel(const float* __restrict__ BB,
                                                    const unsigned int* __restrict__ keybits,
                                                    float* __restrict__ out) {
    __shared__ float         key[NB];
    __shared__ unsigned char visited[NB];
    __shared__ float         redv[32];
    __shared__ int           redi[32];
    __shared__ int           sbest;

    const int t    = threadIdx.x;
    const int lane = t & 31;
    const int wave = t >> 5;

    for (int j = t; j < NB; j += 1024) {
        key[j]     = __uint_as_float(keybits[j]);
        visited[j] = 0;
    }
    __syncthreads();

    float total = 0.0f;

    for (int it = 0; it < NB; it++) {
        // --- per-thread argmin over strided slice (ascending j + strict '<'
        //     keeps the smallest index on ties within a thread) ---
        float bv = __builtin_inff();
        int   bi = 0x7FFFFFFF;
        for (int j = t; j < NB; j += 1024) {
            float v = key[j];
            if (v < bv) { bv = v; bi = j; }
        }
        // --- wave argmin (tie -> smaller index) ---
        for (int off = 16; off > 0; off >>= 1) {
            float ov = __shfl_down(bv, off, 32);
            int   oi = __shfl_down(bi, off, 32);
            if (ov < bv || (ov == bv && oi < bi)) { bv = ov; bi = oi; }
        }
        if (lane == 0) { redv[wave] = bv; redi[wave] = bi; }
        __syncthreads();
        // --- cross-wave argmin in wave 0 ---
        if (wave == 0) {
            bv = redv[lane];
            bi = redi[lane];
            for (int off = 16; off > 0; off >>= 1) {
                float ov = __shfl_down(bv, off, 32);
                int   oi = __shfl_down(bi, off, 32);
                if (ov < bv || (ov == bv && oi < bi)) { bv = ov; bi = oi; }
            }
            if (lane == 0) {
                total      += bv;
                visited[bi] = 1;
                key[bi]     = __builtin_inff();
                sbest       = bi;
            }
        }
        __syncthreads();
        // --- relax keys with the freshly visited row (coalesced float4) ---
        const float4 d = ((const float4*)(BB + (size_t)sbest * NB))[t];
        const int j    = t << 2;
        if (!visited[j    ]) key[j    ] = fminf(key[j    ], d.x);
        if (!visited[j + 1]) key[j + 1] = fminf(key[j + 1], d.y);
        if (!visited[j + 2]) key[j + 2] = fminf(key[j + 2], d.z);
        if (!visited[j + 3]) key[j + 3] = fminf(key[j + 3], d.w);
        __syncthreads();
    }

    if (t == 0) out[0] = total;
}

// ---------------------------------------------------------------------------
// Launch
// ---------------------------------------------------------------------------
extern "C" void kernel_launch(void* const* d_in, const int* in_sizes, int n_in,
                              void* d_out, int out_size, void* d_ws, size_t ws_size,
                              hipStream_t stream) {
    const float* batch  = (const float*)d_in[0];   // 16384 x 128 (only first 4096 rows used)
    const float* output = (const float*)d_in[1];   // 4096 x 128
    float*       outp   = (float*)d_out;

    char* ws = (char*)d_ws;
    const size_t BLK = (size_t)NA * NB * sizeof(float);        // 64 MB
    float*        AB      = (float*)(ws);                      // A-B distances
    float*        BB      = (float*)(ws + BLK);                // B-B distances
    float*        sqa     = (float*)(ws + 2 * BLK);
    float*        sqb     = (float*)(ws + 2 * BLK + NA * sizeof(float));
    unsigned int* keybits = (unsigned int*)(ws + 2 * BLK + (NA + NB) * sizeof(float));

    // squared norms (8 waves/block, 1 point per wave)
    sqnorm_kernel<<<NA / 8, 256, 0, stream>>>(batch,  sqa, NA);
    sqnorm_kernel<<<NB / 8, 256, 0, stream>>>(output, sqb, NB);

    keyinit_kernel<<<NB / 256, 256, 0, stream>>>(keybits);

    // distance blocks via WMMA: A-B then B-B
    dim3 gg(NB / TN, NA / TM);
    dist_tile_kernel<<<gg, 256, 0, stream>>>(batch,  output, sqa, sqb, AB);
    dist_tile_kernel<<<gg, 256, 0, stream>>>(output, output, sqb, sqb, BB);

    // key0[j] = min_i dist(A_i, B_j)
    dim3 gc(NB / 256, NA / 256);
    colmin_kernel<<<gc, 256, 0, stream>>>(AB, keybits);

    // serial Prim over contracted graph
    prim_kernel<<<1, 1024, 0, stream>>>(BB, keybits, outp);
}